// SoftPhongShaderShadow_18365280157914
// MI455X (gfx1250) — compile-verified
//
#include <hip/hip_runtime.h>
#include <stdint.h>

#define SIGMA_F 1.0e-4f
#define GAMMA_F 1.0e-4f
#define ZNEARF  1.0f
#define ZFARF   100.0f
#define EPSF    1e-10f
#define KFRAG   8
#define TILE    256

// LDS layout (bytes) for one 256-pixel tile
#define OFF_P2F    0u       // 256*8*4  = 8192
#define OFF_BARY   8192u    // 256*24*4 = 24576
#define OFF_ZBUF   32768u   // 8192
#define OFF_DST    40960u   // 8192
#define OFF_VIS    49152u   // 1024
#define SMEM_BYTES 50176u

#if defined(__HIP_DEVICE_COMPILE__) && defined(__gfx1250__) && \
    __has_builtin(__builtin_amdgcn_tensor_load_to_lds) &&      \
    __has_builtin(__builtin_amdgcn_s_wait_tensorcnt)
#define USE_TDM 1
#else
#define USE_TDM 0
#endif

#if USE_TDM
typedef __attribute__((ext_vector_type(4))) unsigned int u32x4;
typedef __attribute__((ext_vector_type(8))) int          i32x8;
typedef __attribute__((ext_vector_type(4))) int          i32x4;

// 1-D contiguous global->LDS tile copy via the Tensor Data Mover.
// D# per CDNA5 ISA ch.8: group0 = {count/flags, lds_addr, global_addr, type=2},
// group1 = {data_size=4B, tensor_dim0=n, tensor_dim1=1, tile_dim0=n, tile_dim1=1,
//           tensor_dim0_stride=n}. Groups 2/3 zero (dims 2+ unused).
// This toolchain's builtin is the 6-arg form: the 5th operand (int32x8) is
// passed zero-filled; cpol = 0.
__device__ __forceinline__ void tdm_copy_1d(unsigned lds_byte_addr,
                                            const void* gsrc,
                                            unsigned nelem /* 4B units, <=65535 */) {
  unsigned long long ga = (unsigned long long)(uintptr_t)gsrc;
  u32x4 g0;
  g0[0] = 1u;                                                   // count=1 (valid), user mode
  g0[1] = lds_byte_addr;                                        // lds_addr
  g0[2] = (unsigned)ga;                                         // global_addr[31:0]
  g0[3] = (unsigned)((ga >> 32) & 0x01FFFFFFull) | (2u << 30);  // addr[56:32] | type=2
  i32x8 g1;
  g1[0] = (int)(2u << 16);                                      // data_size = 4 bytes
  g1[1] = (int)((nelem & 0xFFFFu) << 16);                       // tensor_dim0[15:0]
  g1[2] = (int)(((nelem >> 16) & 0xFFFFu) | (1u << 16));        // tensor_dim0[31:16] | tensor_dim1=1
  g1[3] = (int)(nelem << 16);                                   // tile_dim0
  g1[4] = 1;                                                    // tile_dim1=1, tile_dim2=0
  g1[5] = (int)nelem;                                           // tensor_dim0_stride[31:0]
  g1[6] = 0;
  g1[7] = 0;
  i32x4 g2 = {0, 0, 0, 0};
  i32x4 g3 = {0, 0, 0, 0};
  i32x8 g4 = {0, 0, 0, 0, 0, 0, 0, 0};
  __builtin_amdgcn_tensor_load_to_lds(g0, g1, g2, g3, g4, 0);
}
#endif

__global__ __launch_bounds__(TILE) void sps_kernel(
    const int*   __restrict__ p2f,   const float* __restrict__ bary,
    const float* __restrict__ zbuf,  const float* __restrict__ dists,
    const float* __restrict__ verts, const int*   __restrict__ faces,
    const float* __restrict__ vnorm, const float* __restrict__ vcol,
    const float* __restrict__ vis,
    const float* __restrict__ Lp, const float* __restrict__ La,
    const float* __restrict__ Ld, const float* __restrict__ Ls,
    const float* __restrict__ cam, const float* __restrict__ Ma,
    const float* __restrict__ Md, const float* __restrict__ Ms,
    float* __restrict__ out, int npix)
{
  __shared__ __align__(16) unsigned char smem[SMEM_BYTES];
  const unsigned t    = threadIdx.x;
  const unsigned pix0 = blockIdx.x * TILE;

#if USE_TDM
  if (t < 32u) {  // wave 0 only issues the DMA (TDM ignores EXEC; once per wave)
    const unsigned sbase = (unsigned)(uintptr_t)(void*)smem;  // LDS offset = low 32 bits
    tdm_copy_1d(sbase + OFF_P2F,  p2f   + (size_t)pix0 * 8u,  TILE * 8u);
    tdm_copy_1d(sbase + OFF_BARY, bary  + (size_t)pix0 * 24u, TILE * 24u);
    tdm_copy_1d(sbase + OFF_ZBUF, zbuf  + (size_t)pix0 * 8u,  TILE * 8u);
    tdm_copy_1d(sbase + OFF_DST,  dists + (size_t)pix0 * 8u,  TILE * 8u);
    tdm_copy_1d(sbase + OFF_VIS,  vis   + (size_t)pix0,       TILE);
    __builtin_amdgcn_s_wait_tensorcnt(0);
  }
  __syncthreads();
#else
  {
    int* d0 = (int*)(smem + OFF_P2F);
    const int* s0 = p2f + (size_t)pix0 * 8u;
    for (unsigned i = t; i < TILE * 8u; i += TILE) d0[i] = s0[i];
    float* d1 = (float*)(smem + OFF_BARY);
    const float* s1 = bary + (size_t)pix0 * 24u;
    for (unsigned i = t; i < TILE * 24u; i += TILE) d1[i] = s1[i];
    float* d2 = (float*)(smem + OFF_ZBUF);
    const float* s2 = zbuf + (size_t)pix0 * 8u;
    for (unsigned i = t; i < TILE * 8u; i += TILE) d2[i] = s2[i];
    float* d3 = (float*)(smem + OFF_DST);
    const float* s3 = dists + (size_t)pix0 * 8u;
    for (unsigned i = t; i < TILE * 8u; i += TILE) d3[i] = s3[i];
    float* d4 = (float*)(smem + OFF_VIS);
    const float* s4 = vis + (size_t)pix0;
    for (unsigned i = t; i < TILE; i += TILE) d4[i] = s4[i];
  }
  __syncthreads();
#endif

  const int*   sp2f = (const int*)  (smem + OFF_P2F)  + t * 8u;
  const float* sbar = (const float*)(smem + OFF_BARY) + t * 24u;
  const float* szb  = (const float*)(smem + OFF_ZBUF) + t * 8u;
  const float* sds  = (const float*)(smem + OFF_DST)  + t * 8u;
  const float  visv = ((const float*)(smem + OFF_VIS))[t];

  // Uniform scene constants (broadcast loads, L2 hits)
  const float lx = Lp[0],  ly = Lp[1],  lz = Lp[2];
  const float cxx = cam[0], cyy = cam[1], czz = cam[2];
  const float ar = La[0] * Ma[0], ag = La[1] * Ma[1], ab = La[2] * Ma[2];
  const float dr = Ld[0] * Md[0], dg = Ld[1] * Md[1], db = Ld[2] * Md[2];
  const float sr = Ls[0] * Ms[0], sg = Ls[1] * Ms[1], sb2 = Ls[2] * Ms[2];

  float cRk[KFRAG], cGk[KFRAG], cBk[KFRAG], prk[KFRAG], zik[KFRAG];
  float zmax = EPSF;
  float keep = 1.0f;

#pragma unroll
  for (int k = 0; k < KFRAG; ++k) {
    const int f = sp2f[k];
    float r = 0.f, g = 0.f, b = 0.f, p = 0.f, z = 0.f;
    if (f >= 0) {  // masked fragments have exactly-zero blend weight in reference
      const int i0 = faces[3 * f + 0], i1 = faces[3 * f + 1], i2 = faces[3 * f + 2];
      const float b0 = sbar[3 * k + 0], b1 = sbar[3 * k + 1], b2 = sbar[3 * k + 2];
      // interpolate position / normal / color (random L2-resident gathers)
      float px = b0 * verts[3 * i0 + 0] + b1 * verts[3 * i1 + 0] + b2 * verts[3 * i2 + 0];
      float py = b0 * verts[3 * i0 + 1] + b1 * verts[3 * i1 + 1] + b2 * verts[3 * i2 + 1];
      float pz = b0 * verts[3 * i0 + 2] + b1 * verts[3 * i1 + 2] + b2 * verts[3 * i2 + 2];
      float nx = b0 * vnorm[3 * i0 + 0] + b1 * vnorm[3 * i1 + 0] + b2 * vnorm[3 * i2 + 0];
      float ny = b0 * vnorm[3 * i0 + 1] + b1 * vnorm[3 * i1 + 1] + b2 * vnorm[3 * i2 + 1];
      float nz = b0 * vnorm[3 * i0 + 2] + b1 * vnorm[3 * i1 + 2] + b2 * vnorm[3 * i2 + 2];
      float tr = b0 * vcol[3 * i0 + 0] + b1 * vcol[3 * i1 + 0] + b2 * vcol[3 * i2 + 0];
      float tg = b0 * vcol[3 * i0 + 1] + b1 * vcol[3 * i1 + 1] + b2 * vcol[3 * i2 + 1];
      float tb = b0 * vcol[3 * i0 + 2] + b1 * vcol[3 * i1 + 2] + b2 * vcol[3 * i2 + 2];
      // Phong
      float rn = 1.f / fmaxf(sqrtf(nx * nx + ny * ny + nz * nz), 1e-12f);
      nx *= rn; ny *= rn; nz *= rn;
      float ldx = lx - px, ldy = ly - py, ldz = lz - pz;
      float rl = 1.f / fmaxf(sqrtf(ldx * ldx + ldy * ldy + ldz * ldz), 1e-12f);
      ldx *= rl; ldy *= rl; ldz *= rl;
      const float ca  = nx * ldx + ny * ldy + nz * ldz;
      const float dif = fmaxf(ca, 0.f);
      float vx = cxx - px, vy = cyy - py, vz = czz - pz;
      float rv = 1.f / fmaxf(sqrtf(vx * vx + vy * vy + vz * vz), 1e-12f);
      vx *= rv; vy *= rv; vz *= rv;
      const float rfx = 2.f * ca * nx - ldx;
      const float rfy = 2.f * ca * ny - ldy;
      const float rfz = 2.f * ca * nz - ldz;
      float sp = fmaxf(vx * rfx + vy * rfy + vz * rfz, 0.f);
      sp = (ca > 0.f) ? sp : 0.f;
      sp = sp * sp; sp = sp * sp; sp = sp * sp;   // x^8
      sp = sp * sp; sp = sp * sp; sp = sp * sp;   // x^64
      const float dv = dif * visv;
      r = (ar + dr * dv) * tr + sr * sp;
      g = (ag + dg * dv) * tg + sg * sp;
      b = (ab + db * dv) * tb + sb2 * sp;
      // blend terms
      p = 1.f / (1.f + expf(sds[k] * (1.0f / SIGMA_F)));   // sigmoid(-d/sigma)
      z = (ZFARF - szb[k]) * (1.0f / (ZFARF - ZNEARF));
      zmax = fmaxf(zmax, z);
    }
    keep *= (1.f - p);
    cRk[k] = r; cGk[k] = g; cBk[k] = b; prk[k] = p; zik[k] = z;
  }

  const float delta = expf((EPSF - zmax) * (1.0f / GAMMA_F));
  float denom = delta;
  float wn[KFRAG];
#pragma unroll
  for (int k = 0; k < KFRAG; ++k) {
    wn[k] = prk[k] * expf((zik[k] - zmax) * (1.0f / GAMMA_F));
    denom += wn[k];
  }
  const float rden = 1.0f / denom;
  float R = delta * rden, G = delta * rden, B = delta * rden;  // BG = (1,1,1)
#pragma unroll
  for (int k = 0; k < KFRAG; ++k) {
    const float w = wn[k] * rden;
    R += w * cRk[k]; G += w * cGk[k]; B += w * cBk[k];
  }

  const unsigned pix = pix0 + t;
  if ((int)pix < npix) {
    float4 o; o.x = R; o.y = G; o.z = B; o.w = 1.0f - keep;
    ((float4*)out)[pix] = o;
  }
}

extern "C" void kernel_launch(void* const* d_in, const int* in_sizes, int n_in,
                              void* d_out, int out_size, void* d_ws, size_t ws_size,
                              hipStream_t stream) {
  (void)n_in; (void)out_size; (void)d_ws; (void)ws_size;
  const int*   p2f   = (const int*)  d_in[0];
  const float* bary  = (const float*)d_in[1];
  const float* zbuf  = (const float*)d_in[2];
  const float* dists = (const float*)d_in[3];
  const float* verts = (const float*)d_in[4];
  const int*   faces = (const int*)  d_in[5];
  const float* vnorm = (const float*)d_in[6];
  const float* vcol  = (const float*)d_in[7];
  const float* vis   = (const float*)d_in[8];
  const float* Lp    = (const float*)d_in[9];
  const float* La    = (const float*)d_in[10];
  const float* Ld    = (const float*)d_in[11];
  const float* Ls    = (const float*)d_in[12];
  const float* cam   = (const float*)d_in[13];
  const float* Ma    = (const float*)d_in[14];
  const float* Md    = (const float*)d_in[15];
  const float* Ms    = (const float*)d_in[16];

  const int npix = in_sizes[0] / KFRAG;           // N*H*W = 524288 (divisible by TILE)
  const int nblk = (npix + TILE - 1) / TILE;
  sps_kernel<<<dim3(nblk), dim3(TILE), 0, stream>>>(
      p2f, bary, zbuf, dists, verts, faces, vnorm, vcol, vis,
      Lp, La, Ld, Ls, cam, Ma, Md, Ms, (float*)d_out, npix);
}